// GCN_71923522339154
// MI455X (gfx1250) — compile-verified
//
#include <hip/hip_runtime.h>
#include <hip/hip_bf16.h>

#define N_NODES 10000
#define N_EDGES 640000
#define NFEAT   512
#define NHID    128
#define NCLASS  40

typedef float v2f __attribute__((ext_vector_type(2)));
typedef float v8f __attribute__((ext_vector_type(8)));

// ---------------------------------------------------------------- zero fill
__global__ void gcn_zero(float* __restrict__ p, int n) {
    int i = blockIdx.x * blockDim.x + threadIdx.x;
    if (i < n) p[i] = 0.0f;
}

// ---------------------------------------------------------------- GEMM1
// h0[10000,128] = x[10000,512] @ W1[512,128], fp32 WMMA 16x16x4.
// One wave per 16x16 output tile. grid = (625, 8).
__global__ __launch_bounds__(32) void gcn_gemm1(const float* __restrict__ x,
                                                const float* __restrict__ W1,
                                                float* __restrict__ h0) {
    const int lane  = threadIdx.x;
    const int m     = lane & 15;        // row/col within tile
    const int half  = lane >> 4;        // K-half select (kb = 2*half)
    const int row0  = blockIdx.x * 16;  // 625 tiles * 16 = 10000 exactly
    const int col0  = blockIdx.y * 16;  // 8 tiles * 16 = 128

    v8f acc = (v8f)0.0f;
    const float* xrow = x + (size_t)(row0 + m) * NFEAT + half * 2;
    for (int k = 0; k < NFEAT; k += 4) {
        // A 16x4: lane(m,half) holds A[m][k+2*half .. +1]
        v2f a = *(const v2f*)(xrow + k);
        // B 4x16: lane(m,half) holds B[k+2*half][col0+m], B[k+2*half+1][col0+m]
        const float* bp = W1 + (size_t)(k + half * 2) * NHID + col0 + m;
        v2f b;
        b.x = bp[0];
        b.y = bp[NHID];
        acc = __builtin_amdgcn_wmma_f32_16x16x4_f32(false, a, false, b,
                                                    (short)0, acc, false, false);
    }
    // C/D layout: VGPR r -> row r (lanes 0-15) / row 8+r (lanes 16-31), col = m
#pragma unroll
    for (int r = 0; r < 8; ++r) {
        h0[(size_t)(row0 + r + half * 8) * NHID + col0 + m] = acc[r];
    }
}

// ---------------------------------------------------------------- GEMM2
// h2[10000,40] = h[10000,128] @ W2[128,40], N padded 40->48 via zero B cols.
// grid = (625, 3).
__global__ __launch_bounds__(32) void gcn_gemm2(const float* __restrict__ h,
                                                const float* __restrict__ W2,
                                                float* __restrict__ h2) {
    const int lane = threadIdx.x;
    const int m    = lane & 15;
    const int half = lane >> 4;
    const int row0 = blockIdx.x * 16;
    const int col  = blockIdx.y * 16 + m;   // 0..47, valid if < 40

    v8f acc = (v8f)0.0f;
    const float* hrow = h + (size_t)(row0 + m) * NHID + half * 2;
    for (int k = 0; k < NHID; k += 4) {
        v2f a = *(const v2f*)(hrow + k);
        v2f b;
        if (col < NCLASS) {
            const float* bp = W2 + (size_t)(k + half * 2) * NCLASS + col;
            b.x = bp[0];
            b.y = bp[NCLASS];
        } else {
            b.x = 0.0f; b.y = 0.0f;
        }
        acc = __builtin_amdgcn_wmma_f32_16x16x4_f32(false, a, false, b,
                                                    (short)0, acc, false, false);
    }
    if (col < NCLASS) {
#pragma unroll
        for (int r = 0; r < 8; ++r) {
            h2[(size_t)(row0 + r + half * 8) * NCLASS + col] = acc[r];
        }
    }
}

// ---------------------------------------------------------------- SpMM D=128
// s1[dst] += w[e] * h0[src].  2 edges per 256-thread block, 1 thread/feature.
__global__ void gcn_spmm128(const float* __restrict__ h0,
                            const float* __restrict__ ew,
                            const int* __restrict__ esrc,
                            const int* __restrict__ edst,
                            float* __restrict__ s1) {
    int e = blockIdx.x * 2 + (threadIdx.x >> 7);
    int f = threadIdx.x & 127;
    if (e >= N_EDGES) return;
    int s = esrc[e], d = edst[e];
    float v = ew[e] * h0[(size_t)s * NHID + f];
    unsafeAtomicAdd(&s1[(size_t)d * NHID + f], v);
}

// ---------------------------------------------------------------- SpMM D=40
// 64 threads per edge (24 idle lanes), grid-flat indexing.
__global__ void gcn_spmm40(const float* __restrict__ h2,
                           const float* __restrict__ ew,
                           const int* __restrict__ esrc,
                           const int* __restrict__ edst,
                           float* __restrict__ s2) {
    long long gid = (long long)blockIdx.x * blockDim.x + threadIdx.x;
    int e = (int)(gid >> 6);
    int f = (int)(gid & 63);
    if (e >= N_EDGES || f >= NCLASS) return;
    int s = esrc[e], d = edst[e];
    float v = ew[e] * h2[(size_t)s * NCLASS + f];
    unsafeAtomicAdd(&s2[(size_t)d * NCLASS + f], v);
}

// ---------------------------------------------------------------- bias + ReLU
__global__ void gcn_bias_relu(float* __restrict__ s1, const float* __restrict__ b1, int n) {
    int i = blockIdx.x * blockDim.x + threadIdx.x;
    if (i < n) {
        float v = s1[i] + b1[i & (NHID - 1)];
        s1[i] = v > 0.0f ? v : 0.0f;
    }
}

// ---------------------------------------------------------------- bias + log_softmax
// One wave per row (wave32). Lanes 0..31 cover col=lane; lanes 0..7 also col=lane+32.
__global__ void gcn_logsoftmax(const float* __restrict__ s2,
                               const float* __restrict__ b2,
                               float* __restrict__ out) {
    int row  = blockIdx.x * 8 + (threadIdx.x >> 5);
    int lane = threadIdx.x & 31;
    if (row >= N_NODES) return;
    const float* rp = s2 + (size_t)row * NCLASS;

    float lv0 = rp[lane] + b2[lane];
    float lv1 = (lane < NCLASS - 32) ? (rp[lane + 32] + b2[lane + 32]) : -3.402823466e38f;

    float mx = fmaxf(lv0, lv1);
#pragma unroll
    for (int o = 16; o > 0; o >>= 1) mx = fmaxf(mx, __shfl_xor(mx, o, 32));

    float se = __expf(lv0 - mx) + ((lane < NCLASS - 32) ? __expf(lv1 - mx) : 0.0f);
#pragma unroll
    for (int o = 16; o > 0; o >>= 1) se += __shfl_xor(se, o, 32);

    float ls = mx + __logf(se);
    float* op = out + (size_t)row * NCLASS;
    op[lane] = lv0 - ls;
    if (lane < NCLASS - 32) op[lane + 32] = lv1 - ls;
}

// ---------------------------------------------------------------- launch
extern "C" void kernel_launch(void* const* d_in, const int* in_sizes, int n_in,
                              void* d_out, int out_size, void* d_ws, size_t ws_size,
                              hipStream_t stream) {
    const float* x   = (const float*)d_in[0];
    const float* ew  = (const float*)d_in[1];
    const float* W1  = (const float*)d_in[2];
    const float* b1  = (const float*)d_in[3];
    const float* W2  = (const float*)d_in[4];
    const float* b2  = (const float*)d_in[5];
    const int*  esrc = (const int*)d_in[6];
    const int*  edst = (const int*)d_in[7];
    float* out = (float*)d_out;

    // workspace layout (floats)
    float* ws = (float*)d_ws;
    float* h0 = ws;                                   // 10000*128
    float* s1 = h0 + (size_t)N_NODES * NHID;          // 10000*128 (spmm1 acc -> h in place)
    float* h2 = s1 + (size_t)N_NODES * NHID;          // 10000*40
    float* s2 = h2 + (size_t)N_NODES * NCLASS;        // 10000*40

    const int nS1 = N_NODES * NHID;    // 1,280,000
    const int nS2 = N_NODES * NCLASS;  //   400,000

    // zero atomic accumulators (ws is not re-zeroed between graph replays)
    gcn_zero<<<(nS1 + 255) / 256, 256, 0, stream>>>(s1, nS1);
    gcn_zero<<<(nS2 + 255) / 256, 256, 0, stream>>>(s2, nS2);

    // layer 1
    gcn_gemm1<<<dim3(N_NODES / 16, NHID / 16), 32, 0, stream>>>(x, W1, h0);
    gcn_spmm128<<<N_EDGES / 2, 256, 0, stream>>>(h0, ew, esrc, edst, s1);
    gcn_bias_relu<<<(nS1 + 255) / 256, 256, 0, stream>>>(s1, b1, nS1);

    // layer 2
    gcn_gemm2<<<dim3(N_NODES / 16, 3), 32, 0, stream>>>(s1, W2, h2);
    {
        long long total = (long long)N_EDGES * 64;
        int blocks = (int)((total + 255) / 256);
        gcn_spmm40<<<blocks, 256, 0, stream>>>(h2, ew, esrc, edst, s2);
    }

    // bias + log_softmax -> d_out
    gcn_logsoftmax<<<(N_NODES + 7) / 8, 256, 0, stream>>>(s2, b2, out);
}